// MFNetLayerL1_52089363366464
// MI455X (gfx1250) — compile-verified
//
#include <hip/hip_runtime.h>

typedef __attribute__((ext_vector_type(2))) float v2f;
typedef __attribute__((ext_vector_type(8))) float v8f;

#define N_RX 8
#define N_TX 4

// x<0 : x - ln2 ; x>=0 : log(1 - 0.5*exp(-x)); arg of log is in [0.5,1) -> well conditioned
__device__ __forceinline__ float laplace_log_cdf(float x) {
  float e   = __expf(-__builtin_fabsf(x));
  float pos = __logf(__builtin_fmaf(-0.5f, e, 1.0f));
  return (x < 0.0f) ? (x - 0.6931471805599453f) : pos;
}

// {-3,-1,1,3}/sqrt(5); for compile-time i this folds to a literal
__device__ __forceinline__ float sym_of(int i) {
  const float s1 = 0.4472135954999579f;   //  1/sqrt(5)
  const float s3 = 1.3416407864998738f;   //  3/sqrt(5)
  float lo = (i & 1) ? -s1 : -s3;
  float hi = (i & 1) ?  s3 :  s1;
  return (i & 2) ? hi : lo;
}

__device__ __forceinline__ float sel4(float a0, float a1, float a2, float a3, int i) {
  float lo = (i & 1) ? a1 : a0;
  float hi = (i & 1) ? a3 : a2;
  return (i & 2) ? hi : lo;
}

__global__ __launch_bounds__(256) void mfnet_layer_kernel(
    const float* __restrict__ log_qi,
    const float* __restrict__ G,
    const float* __restrict__ sqrt_2rho,
    const float* __restrict__ alpha_p,
    float* __restrict__ out, int nprob)
{
  const int lane = threadIdx.x & 31;
  const int wave = threadIdx.x >> 5;
  const int p0   = blockIdx.x * 16 + wave * 2;     // 2 problems per wave, 16 per block
  if (p0 + 2 > nprob) return;                       // wave-uniform guard (EXEC stays all-ones)

  // ---------------- Phase 1: term = (s2r*G) @ s_all^T via V_WMMA_F32_16X16X4_F32 ----------------
  // A (16x4 f32): M = lane&15 (rows 0-7 -> p0, 8-15 -> p0+1); lanes<16 hold K=0,1; lanes>=16 K=2,3
  const int Mrow = lane & 15;
  const int pa   = p0 + (Mrow >> 3);
  const int r    = Mrow & 7;
  const int koff = (lane < 16) ? 0 : 2;
  const float s2r = sqrt_2rho[pa];
  const float* gp = G + (((size_t)pa * N_RX + r) * N_TX + koff);
  v2f A;
  A.x = s2r * gp[0];
  A.y = s2r * gp[1];

  // Phase-2 lane identity: lanes 0-15 -> p0, lanes 16-31 -> p0+1 (matches D layout)
  const int  pme     = p0 + (lane >> 4);
  const int  n       = lane & 15;                   // low nibble of candidate index k
  const bool lo_half = (lane < 16);

  // B-matrix values for the high half (rows t=2,3) are kt-invariant: sym(n>>2), sym(n&3)
  const float symn2 = sym_of(n >> 2);
  const float symn3 = sym_of(n & 3);

  float logp[16];
#pragma unroll 16
  for (int kt = 0; kt < 16; ++kt) {
    // lanes<16 supply rows t=0,1 -> sym(kt>>2), sym(kt&3): literal constants after unroll
    v2f B;
    B.x = lo_half ? sym_of(kt >> 2) : symn2;
    B.y = lo_half ? sym_of(kt & 3)  : symn3;
    v8f C = {};
    v8f D = __builtin_amdgcn_wmma_f32_16x16x4_f32(false, A, false, B, (short)0, C,
                                                  false, false);
    float s = 0.0f;
#pragma unroll
    for (int v = 0; v < 8; ++v) s += laplace_log_cdf(D[v]);
    logp[kt] = s;   // log_p for (pme, k = 16*kt + n)
  }

  // ---------------- Phase 2: sequential xi loop on per-problem 4x4 log_qi state ----------------
  float lq[16];
  {
    const float4* lqs = reinterpret_cast<const float4*>(log_qi) + (size_t)pme * 4;
#pragma unroll
    for (int i = 0; i < 4; ++i) {
      float4 v = lqs[i];
      lq[i*4+0] = v.x; lq[i*4+1] = v.y; lq[i*4+2] = v.z; lq[i*4+3] = v.w;
    }
  }

  const float alpha = alpha_p[0];
  const float na    = 1.0f - alpha;
  const int i2lane  = n >> 2;   // symbol index of TX2 for this lane's k's (lane-constant)
  const int i3lane  = n & 3;    // symbol index of TX3 (lane-constant)

#pragma unroll
  for (int xi = 0; xi < 4; ++xi) {
    // softmax of each row of log_qi (fast exp + fast rcp); row xi is dead and DCE'd
    float q[4][4];
#pragma unroll
    for (int t = 0; t < 4; ++t) {
      float e0 = __expf(lq[t*4+0]);
      float e1 = __expf(lq[t*4+1]);
      float e2 = __expf(lq[t*4+2]);
      float e3 = __expf(lq[t*4+3]);
      float inv = __builtin_amdgcn_rcpf((e0 + e1) + (e2 + e3));
      q[t][0] = e0*inv; q[t][1] = e1*inv; q[t][2] = e2*inv; q[t][3] = e3*inv;
    }
    const float q2v = sel4(q[2][0], q[2][1], q[2][2], q[2][3], i2lane);
    const float q3v = sel4(q[3][0], q[3][1], q[3][2], q[3][3], i3lane);

    float acc[4] = {0.f, 0.f, 0.f, 0.f};
    float accs   = 0.f;
#pragma unroll
    for (int kt = 0; kt < 16; ++kt) {
      const int i0 = kt >> 2;      // compile-time after unroll
      const int i1 = kt & 3;       // compile-time after unroll
      float w;
      if (xi == 0)      w = q[1][i1] * q2v * q3v;
      else if (xi == 1) w = q[0][i0] * q2v * q3v;
      else if (xi == 2) w = q[0][i0] * q[1][i1] * q3v;
      else              w = q[0][i0] * q[1][i1] * q2v;
      const float c = w * logp[kt];
      if (xi == 0)      acc[i0] += c;   // static index
      else if (xi == 1) acc[i1] += c;   // static index
      else              accs += c;      // bin is lane-constant, spread below
    }
    if (xi >= 2) {
      const int bin = (xi == 2) ? i2lane : i3lane;
#pragma unroll
      for (int s = 0; s < 4; ++s) acc[s] = (bin == s) ? accs : 0.0f;
    }

    // reduce the 16 lanes belonging to this problem (xor masks < 16 stay in-half)
#pragma unroll
    for (int m = 1; m < 16; m <<= 1) {
#pragma unroll
      for (int s = 0; s < 4; ++s) acc[s] += __shfl_xor(acc[s], m, 32);
    }

    // blended update of row xi, then per-row max normalization of all rows
#pragma unroll
    for (int s = 0; s < 4; ++s) lq[xi*4+s] = na * lq[xi*4+s] + alpha * acc[s];
#pragma unroll
    for (int t = 0; t < 4; ++t) {
      float m01 = fmaxf(lq[t*4+0], lq[t*4+1]);
      float m23 = fmaxf(lq[t*4+2], lq[t*4+3]);
      float mx  = fmaxf(m01, m23);
#pragma unroll
      for (int s = 0; s < 4; ++s) lq[t*4+s] -= mx;
    }
  }

  // ---------------- Output: one lane per problem writes its 16 floats as 4x float4 ----------------
  if (n == 0) {
    float4* o = reinterpret_cast<float4*>(out) + (size_t)pme * 4;
    o[0] = make_float4(lq[0],  lq[1],  lq[2],  lq[3]);
    o[1] = make_float4(lq[4],  lq[5],  lq[6],  lq[7]);
    o[2] = make_float4(lq[8],  lq[9],  lq[10], lq[11]);
    o[3] = make_float4(lq[12], lq[13], lq[14], lq[15]);
  }
}

extern "C" void kernel_launch(void* const* d_in, const int* in_sizes, int n_in,
                              void* d_out, int out_size, void* d_ws, size_t ws_size,
                              hipStream_t stream) {
  const float* log_qi    = (const float*)d_in[0];   // (N,4,4) f32
  const float* G         = (const float*)d_in[1];   // (N,8,4) f32
  const float* sqrt_2rho = (const float*)d_in[2];   // (N,)    f32
  // d_in[3] = n_var : unused by the reference computation
  const float* alpha_p   = (const float*)d_in[4];   // scalar f32

  float* out = (float*)d_out;
  const int nprob = in_sizes[2];                    // N = 32768
  const int blocks = (nprob + 15) / 16;             // 16 problems per 256-thread block

  mfnet_layer_kernel<<<blocks, 256, 0, stream>>>(log_qi, G, sqrt_2rho, alpha_p, out, nprob);
}